// GCNLayer_54185307407137
// MI455X (gfx1250) — compile-verified
//
#include <hip/hip_runtime.h>

// SpMM via sorted-row tiling + V_WMMA_F32_16X16X4_F32 segment-sum.
// out[r] = sum_{e: row[e]==r} vals[e] * embeds[col[e]]
//
// Kernel 1: parallel binary search of all 3126 tile edge-boundaries into d_ws
//           (removes ~34 serially-dependent loads from every main block).
// Kernel 2: each block owns 16 output rows; edges [bnd[t], bnd[t+1]).
//   Segment-sum as WMMA chain: A[r,k] = (row==base+r)?val:0, B = gathered
//   features. 4 waves = 4 feature slices of 16. Metadata staged per 128-edge
//   chunk in LDS packed as {(col<<5)|local_row, val_bits}; invalid slots get
//   local_row=31 (never matches r in [0,16)) so tail masking is free. The two
//   K values a lane needs are adjacent edges -> one aligned ds_load_b128 per
//   WMMA group. Two accumulators break the WMMA RAW chain. No atomics; every
//   output element written exactly once.

typedef __attribute__((ext_vector_type(2))) float v2f;
typedef __attribute__((ext_vector_type(8))) float v8f;

#define N_NODES 50000
#define N_EDGES 800000
#define D_FEAT 64
#define ROWS_PER_TILE 16
#define N_TILES (N_NODES / ROWS_PER_TILE)   // 3125
#define CHUNK 128
#define BLOCK_THREADS 128

__device__ __forceinline__ int lower_bound_i32(const int* __restrict__ arr,
                                               int n, int key) {
    int lo = 0, hi = n;
    while (lo < hi) {
        int mid = (lo + hi) >> 1;
        if (arr[mid] < key) lo = mid + 1; else hi = mid;
    }
    return lo;
}

// ---- kernel 1: tile boundary precompute (3126 independent searches) ----
__global__ __launch_bounds__(256) void build_tile_bounds(
    const int* __restrict__ row_idx, int* __restrict__ bnd) {
    const int i = blockIdx.x * blockDim.x + threadIdx.x;
    if (i <= N_TILES)
        bnd[i] = lower_bound_i32(row_idx, N_EDGES, i * ROWS_PER_TILE);
}

// ---- kernel 2: WMMA segment-sum SpMM ----
__global__ __launch_bounds__(BLOCK_THREADS) void gcn_spmm_wmma(
    const int* __restrict__ row_idx,
    const int* __restrict__ col_idx,
    const float* __restrict__ vals,
    const float* __restrict__ embeds,
    const int* __restrict__ bnd,
    float* __restrict__ out) {

    // {packed, val_bits} per edge; uint4 view needs 16B alignment.
    __shared__ __align__(16) unsigned int meta[CHUNK * 2];

    const int tid  = threadIdx.x;
    const int tile = blockIdx.x;
    const int base_row = tile * ROWS_PER_TILE;
    const int wave = tid >> 5;           // 0..3 -> feature slice
    const int lane = tid & 31;
    const int half = lane >> 4;          // lanes 16-31 -> second K/M group
    const int r    = lane & 15;          // A row / B col / C col
    const int fr   = wave * 16 + r;      // feature index for this lane

    const int e_start = bnd[tile];
    const int e_end   = bnd[tile + 1];

    v8f acc0 = {};
    v8f acc1 = {};

    for (int c0 = e_start; c0 < e_end; c0 += CHUNK) {
        // ---- stage chunk metadata (coalesced, once per block) ----
        {
            const int e  = c0 + tid;
            const int es = (e < e_end) ? e : (e_end - 1);   // clamped address
            // Unconditional loads (no EXEC predication), then select.
            const int      rv = row_idx[es];
            const unsigned cv = (unsigned)col_idx[es];      // < 50000, 16 bits
            const unsigned vb = __float_as_uint(vals[es]);
            const unsigned lr = (e < e_end) ? (unsigned)(rv - base_row)  // 0..15
                                            : 31u;          // never matches r
            *(uint2*)&meta[2 * tid] = make_uint2((cv << 5) | lr, vb);
        }
        __syncthreads();

        const int edges_here = (e_end - c0 < CHUNK) ? (e_end - c0) : CHUNK;
        int ng = (edges_here + 3) >> 2;   // groups of K=4
        ng = (ng + 1) & ~1;               // even: padding slots are inert

        for (int g = 0; g < ng; g += 2) {
            // ---- group g -> acc0 ----
            {
                // edges g*4 + 2*half + {0,1}: adjacent -> one b128 LDS read
                const uint4 m = ((const uint4*)meta)[g * 2 + half];
                v2f a, b;
                a[0] = ((m.x & 31u) == (unsigned)r) ? __uint_as_float(m.y) : 0.0f;
                a[1] = ((m.z & 31u) == (unsigned)r) ? __uint_as_float(m.w) : 0.0f;
                b[0] = embeds[(m.x >> 5) * D_FEAT + fr];
                b[1] = embeds[(m.z >> 5) * D_FEAT + fr];
                acc0 = __builtin_amdgcn_wmma_f32_16x16x4_f32(
                    false, a, false, b, (short)0, acc0, false, false);
            }
            // ---- group g+1 -> acc1 (independent of acc0 chain) ----
            {
                const uint4 m = ((const uint4*)meta)[(g + 1) * 2 + half];
                v2f a, b;
                a[0] = ((m.x & 31u) == (unsigned)r) ? __uint_as_float(m.y) : 0.0f;
                a[1] = ((m.z & 31u) == (unsigned)r) ? __uint_as_float(m.w) : 0.0f;
                b[0] = embeds[(m.x >> 5) * D_FEAT + fr];
                b[1] = embeds[(m.z >> 5) * D_FEAT + fr];
                acc1 = __builtin_amdgcn_wmma_f32_16x16x4_f32(
                    false, a, false, b, (short)0, acc1, false, false);
            }
        }
        __syncthreads();   // protect LDS before next chunk's staging
    }

    // C layout: lane l, VGPR v -> M = v + 8*half, N = r.
#pragma unroll
    for (int v = 0; v < 8; ++v) {
        const int m = v + 8 * half;
        out[(base_row + m) * D_FEAT + wave * 16 + r] = acc0[v] + acc1[v];
    }
}

extern "C" void kernel_launch(void* const* d_in, const int* in_sizes, int n_in,
                              void* d_out, int out_size, void* d_ws, size_t ws_size,
                              hipStream_t stream) {
    (void)in_sizes; (void)n_in; (void)out_size; (void)ws_size;
    const int*   row_idx = (const int*)d_in[0];
    const int*   col_idx = (const int*)d_in[1];
    const float* vals    = (const float*)d_in[2];
    const float* embeds  = (const float*)d_in[3];
    float*       out     = (float*)d_out;
    int*         bnd     = (int*)d_ws;   // 3126 ints of scratch

    // 1) parallel tile-boundary search (3126 threads)
    build_tile_bounds<<<(N_TILES + 1 + 255) / 256, 256, 0, stream>>>(row_idx, bnd);

    // 2) WMMA segment-sum SpMM
    dim3 grid(N_TILES);                  // 3125 tiles, each owns 16 rows
    dim3 block(BLOCK_THREADS);           // 4 waves = 4 feature slices of 16
    gcn_spmm_wmma<<<grid, block, 0, stream>>>(row_idx, col_idx, vals, embeds,
                                              bnd, out);
}